// HetegModel_29317446762540
// MI455X (gfx1250) — compile-verified
//
#include <hip/hip_runtime.h>
#include <hip/hip_bf16.h>
#include <stddef.h>

// ---------------- problem constants (match reference) ----------------
#define NW 150000
#define NE 30000
#define EP 2000000
#define EE 2000000
#define DF 768
#define DIN 769
#define DH 32
#define DO 32

typedef __attribute__((ext_vector_type(2))) float v2f;
typedef __attribute__((ext_vector_type(8))) float v8f;
typedef __attribute__((ext_vector_type(4))) int   v4i;

#define AS1 __attribute__((address_space(1)))
#define AS3 __attribute__((address_space(3)))

#if defined(__HIP_DEVICE_COMPILE__) && __has_builtin(__builtin_amdgcn_global_load_async_to_lds_b128)
#define PROJ_ASYNC 1
#else
#define PROJ_ASYNC 0
#endif

// ---------------- utility kernels ----------------

__global__ void zero_f32(float* __restrict__ p, size_t n) {
    size_t i = (size_t)blockIdx.x * blockDim.x + threadIdx.x;
    size_t stride = (size_t)gridDim.x * blockDim.x;
    for (; i < n; i += stride) p[i] = 0.0f;
}

// histogram: deg[idx[i]] += 1
__global__ void hist1(const int* __restrict__ idx, int n, float* __restrict__ deg) {
    int i = blockIdx.x * blockDim.x + threadIdx.x;
    if (i < n) atomicAdd(&deg[idx[i]], 1.0f);
}

// ---------------- WMMA projection: h[N,32] = rsqrt(max(deg,1)) * (x @ W) ----------------
// x = concat(seed, emb) (DIN=769).  W is [769][32] row-major.
// Block = 128 threads (4 waves). Block covers 64 rows; wave w owns rows [w*16, w*16+16).
// K loop over emb dims (768 = 192 groups of 4); seed column + row scaling in epilogue.
// Raw emb tiles are DMA'd to LDS with GLOBAL_LOAD_ASYNC_TO_LDS_B128, double-buffered so
// the next chunk's HBM traffic overlaps the current chunk's 32 v_wmma_f32_16x16x4_f32.

#define KC 64                 // k-chunk staged in LDS
#define XPITCH 68             // 64 + 4 pad: A-fragment reads hit distinct banks
#define XTILE (64 * XPITCH)
#define NCHUNK (DF / KC)      // 12

__launch_bounds__(128)
__global__ void project_wmma(const float* __restrict__ emb,
                             const float* __restrict__ seed,
                             const float* __restrict__ deg_out,
                             const float* __restrict__ W,   // [DIN][32]
                             float* __restrict__ h,         // [N][32]
                             int N)
{
    __shared__ float Xs[2 * XTILE];
    __shared__ float rsL[64];

    const int t    = threadIdx.x;
    const int wave = t >> 5;
    const int lane = t & 31;
    const int half = lane >> 4;   // 0: lanes 0-15, 1: lanes 16-31
    const int m    = lane & 15;
    const int row0 = blockIdx.x * 64;

    if (t < 64) {
        int gr = row0 + t;
        float d = (gr < N) ? deg_out[gr] : 1.0f;
        rsL[t] = rsqrtf(fmaxf(d, 1.0f));
    }

    // stage one raw 64x64 f32 emb tile into LDS buffer `buf` (8 b128 ops per thread)
    auto stage = [&](int kb, int buf) {
#pragma unroll
        for (int it = 0; it < 8; ++it) {
            int idx = it * 128 + t;      // 0..1023 -> 64 rows x 16 quads
            int row = idx >> 4;
            int q   = idx & 15;
            int gr  = row0 + row;
            if (gr >= N) gr = N - 1;     // clamp: stays in-bounds; row never stored
            const float* gp = emb + (size_t)gr * DF + kb + q * 4;
            float*       lp = Xs + buf * XTILE + row * XPITCH + q * 4;
#if PROJ_ASYNC
            __builtin_amdgcn_global_load_async_to_lds_b128(
                (AS1 v4i*)gp, (AS3 v4i*)lp, 0, 0);
#else
            *(float4*)lp = *(const float4*)gp;
#endif
        }
    };

    v8f c0 = {};   // cols 0..15
    v8f c1 = {};   // cols 16..31

    stage(0, 0);   // prefetch first chunk

    for (int i = 0; i < NCHUNK; ++i) {
        const int cur = i & 1;
        if (i + 1 < NCHUNK) {
            stage((i + 1) * KC, cur ^ 1);            // overlap next DMA with compute
            asm volatile("s_wait_asynccnt 8" ::: "memory");   // current chunk retired
        } else {
            asm volatile("s_wait_asynccnt 0" ::: "memory");
        }
        __syncthreads();                              // all waves see buf[cur]

        const int kb = i * KC;
        const float* Xw = Xs + cur * XTILE + (wave * 16 + m) * XPITCH;
#pragma unroll
        for (int kg = 0; kg < KC; kg += 4) {
            int kl = kg + half * 2;
            // A 16x4 f32 fragment: lane(half,m): v0=A[m][half*2], v1=A[m][half*2+1]
            v2f a; a.x = Xw[kl]; a.y = Xw[kl + 1];
            // B 4x16 f32 fragment: v0=B[half*2][col], v1=B[half*2+1][col]
            int gk = kb + kl;                        // emb dim -> W row gk+1 (row 0 = seed)
            const float* Wp = W + (size_t)(gk + 1) * DH;
            v2f b0; b0.x = Wp[m];      b0.y = Wp[DH + m];
            v2f b1; b1.x = Wp[16 + m]; b1.y = Wp[DH + 16 + m];
            c0 = __builtin_amdgcn_wmma_f32_16x16x4_f32(false, a, false, b0,
                                                       (short)0, c0, false, false);
            c1 = __builtin_amdgcn_wmma_f32_16x16x4_f32(false, a, false, b1,
                                                       (short)0, c1, false, false);
        }
        __syncthreads();                              // free buf[cur] for next DMA
    }

    // ---- epilogue: row scale by rsqrt(deg), add seed column (k=0), store ----
    // C/D layout: VGPR r: lanes 0-15 -> row r, lanes 16-31 -> row r+8; col = m (+16 for c1)
    const float w0a = W[m];        // W[0][m]
    const float w0b = W[16 + m];   // W[0][16+m]
#pragma unroll
    for (int r = 0; r < 8; ++r) {
        int lrow = wave * 16 + r + half * 8;
        int gr = row0 + lrow;
        if (gr < N) {
            float rs = rsL[lrow];
            float sv = seed[gr] * rs;
            h[(size_t)gr * DH + m]      = rs * c0[r] + sv * w0a;
            h[(size_t)gr * DH + 16 + m] = rs * c1[r] + sv * w0b;
        }
    }
}

// ---------------- edge gather + scatter-add: agg[dst] += h[src] ----------------
// one lane per (edge, dim); h/agg stay resident in the 192MB L2
__global__ void edge_agg(const int* __restrict__ src, const int* __restrict__ dst,
                         const float* __restrict__ h, float* __restrict__ agg,
                         size_t total /* E*32 */) {
    size_t i = (size_t)blockIdx.x * blockDim.x + threadIdx.x;
    if (i >= total) return;
    size_t e = i >> 5;
    int    c = (int)(i & 31);
    int s = src[e];
    int d = dst[e];
    atomicAdd(&agg[(size_t)d * DH + c], h[(size_t)s * DH + c]);
}

// ---------------- relu(agg*rsqrt(deg_in)+b) summed into pooled1[32] ----------------
__launch_bounds__(256)
__global__ void finalize_pool1(const float* __restrict__ agg,
                               const float* __restrict__ deg_in,
                               const float* __restrict__ bias,
                               int n, float* __restrict__ pooled1) {
    __shared__ float red[256];
    const int t = threadIdx.x;
    const int c = t & 31;                 // column stays fixed: stride % 32 == 0
    const float b = bias[c];
    size_t total = (size_t)n * DH;
    size_t stride = (size_t)gridDim.x * 256;
    float acc = 0.0f;
    for (size_t i = (size_t)blockIdx.x * 256 + t; i < total; i += stride) {
        size_t node = i >> 5;
        float v = agg[i] * rsqrtf(fmaxf(deg_in[node], 1.0f)) + b;
        acc += fmaxf(v, 0.0f);
    }
    red[t] = acc;
    __syncthreads();
    if (t < 32) {
        float s = 0.0f;
#pragma unroll
        for (int j = 0; j < 8; ++j) s += red[c + 32 * j];
        atomicAdd(&pooled1[c], s);
    }
}

// ---------------- column sums of emb into pooled0[1..768] ----------------
__global__ void colsum(const float* __restrict__ emb, int nrows, float* __restrict__ out) {
    int col = blockIdx.x * blockDim.x + threadIdx.x;   // gridDim.x * 256 == 768
    float s = 0.0f;
    for (int r = blockIdx.y; r < nrows; r += gridDim.y)
        s += emb[(size_t)r * DF + col];
    atomicAdd(&out[col], s);
}

// ---------------- scalar sum (seeds) into pooled0[0] ----------------
__global__ void sum_atomic(const float* __restrict__ v, int n, float* __restrict__ out) {
    __shared__ float red[256];
    float s = 0.0f;
    for (size_t i = (size_t)blockIdx.x * 256 + threadIdx.x; i < (size_t)n;
         i += (size_t)gridDim.x * 256)
        s += v[i];
    red[threadIdx.x] = s;
    __syncthreads();
    for (int st = 128; st > 0; st >>= 1) {
        if (threadIdx.x < st) red[threadIdx.x] += red[threadIdx.x + st];
        __syncthreads();
    }
    if (threadIdx.x == 0) atomicAdd(out, red[0]);
}

// ---------------- final score: [1,32] ----------------
__global__ void score_k(const float* __restrict__ pooled0, const float* __restrict__ pooled1,
                        const float* __restrict__ lin0W, const float* __restrict__ lin0b,
                        const float* __restrict__ lin1W, const float* __restrict__ lin1b,
                        float* __restrict__ out) {
    int c = threadIdx.x;
    if (c >= DO) return;
    float s = lin0b[c] + lin1b[c];
    for (int k = 0; k < DIN; ++k) s += pooled0[k] * lin0W[k * DO + c];
    for (int j = 0; j < DH;  ++j) s += pooled1[j] * lin1W[j * DO + c];
    out[c] = s;
}

// ---------------- launch ----------------
extern "C" void kernel_launch(void* const* d_in, const int* in_sizes, int n_in,
                              void* d_out, int out_size, void* d_ws, size_t ws_size,
                              hipStream_t stream) {
    const float* emb_word   = (const float*)d_in[0];
    const float* emb_emoji  = (const float*)d_in[1];
    const float* seed_word  = (const float*)d_in[2];
    const float* seed_emoji = (const float*)d_in[3];
    const int*   post_src   = (const int*)d_in[4];
    const int*   post_dst   = (const int*)d_in[5];
    const int*   emoji_src  = (const int*)d_in[6];
    const int*   emoji_dst  = (const int*)d_in[7];
    const float* W_post     = (const float*)d_in[8];
    const float* b_post     = (const float*)d_in[9];
    const float* W_emoji    = (const float*)d_in[10];
    const float* b_emoji    = (const float*)d_in[11];
    const float* lin0_W     = (const float*)d_in[12];
    const float* lin0_b     = (const float*)d_in[13];
    const float* lin1_W     = (const float*)d_in[14];
    const float* lin1_b     = (const float*)d_in[15];
    float* out = (float*)d_out;

    // workspace layout: atomically-accumulated regions first (zeroed every call)
    float* ws = (float*)d_ws;
    size_t o = 0;
    float* deg_out_word  = ws + o; o += NW;
    float* deg_in_word   = ws + o; o += NW;
    float* deg_out_emoji = ws + o; o += NE;
    float* deg_in_emoji  = ws + o; o += NE;
    float* agg_word      = ws + o; o += (size_t)NW * DH;
    float* agg_emoji     = ws + o; o += (size_t)NE * DH;
    float* pooled0       = ws + o; o += DIN;
    float* pooled1       = ws + o; o += DO;
    size_t zero_n = o;
    float* h_word        = ws + o; o += (size_t)NW * DH;
    float* h_emoji       = ws + o; o += (size_t)NE * DH;

    // 1) clear accumulators
    zero_f32<<<dim3(2048), dim3(256), 0, stream>>>(ws, zero_n);

    // 2) degrees (float histograms)
    hist1<<<(EP + 255) / 256, 256, 0, stream>>>(post_src,  EP, deg_out_word);
    hist1<<<(EP + 255) / 256, 256, 0, stream>>>(post_dst,  EP, deg_in_emoji);
    hist1<<<(EE + 255) / 256, 256, 0, stream>>>(emoji_src, EE, deg_out_emoji);
    hist1<<<(EE + 255) / 256, 256, 0, stream>>>(emoji_dst, EE, deg_in_word);

    // 3) WMMA projections (async-LDS double-buffered; f32 WMMA at the BW roof)
    project_wmma<<<(NW + 63) / 64, 128, 0, stream>>>(emb_word,  seed_word,  deg_out_word,
                                                     W_post,  h_word,  NW);
    project_wmma<<<(NE + 63) / 64, 128, 0, stream>>>(emb_emoji, seed_emoji, deg_out_emoji,
                                                     W_emoji, h_emoji, NE);

    // 4) edge gather/scatter (L2-resident atomics)
    {
        size_t tp = (size_t)EP * DH;
        size_t te = (size_t)EE * DH;
        edge_agg<<<(unsigned)((tp + 255) / 256), 256, 0, stream>>>(post_src,  post_dst,
                                                                   h_word,  agg_emoji, tp);
        edge_agg<<<(unsigned)((te + 255) / 256), 256, 0, stream>>>(emoji_src, emoji_dst,
                                                                   h_emoji, agg_word, te);
    }

    // 5) relu + sum-pool into pooled1  (post relation -> emoji nodes uses b_post)
    finalize_pool1<<<1024, 256, 0, stream>>>(agg_emoji, deg_in_emoji, b_post,  NE, pooled1);
    finalize_pool1<<<1024, 256, 0, stream>>>(agg_word,  deg_in_word,  b_emoji, NW, pooled1);

    // 6) pooled0 = sum of concat(seed, emb) over all nodes
    colsum<<<dim3(DF / 256, 512), 256, 0, stream>>>(emb_word,  NW, pooled0 + 1);
    colsum<<<dim3(DF / 256, 512), 256, 0, stream>>>(emb_emoji, NE, pooled0 + 1);
    sum_atomic<<<256, 256, 0, stream>>>(seed_word,  NW, pooled0);
    sum_atomic<<<256, 256, 0, stream>>>(seed_emoji, NE, pooled0);

    // 7) final score [1,32]
    score_k<<<1, 32, 0, stream>>>(pooled0, pooled1, lin0_W, lin0_b, lin1_W, lin1_b, out);
}